// RayTracingModule_38104949850632
// MI455X (gfx1250) — compile-verified
//
#include <hip/hip_runtime.h>
#include <math.h>

typedef float v2f __attribute__((ext_vector_type(2)));
typedef float v8f __attribute__((ext_vector_type(8)));

#define HH 1024
#define WW 1024
#define NPIX (HH*WW)

// LDS staging layout (dwords): centers[48] | radii[16] | colors[48] | lp[3] | lc[3]
#define STG_CEN 0
#define STG_RAD 48
#define STG_COL 64
#define STG_LP  112
#define STG_LC  115
#define STG_N   118

__global__ __launch_bounds__(256) void raytrace_wmma_kernel(
    const float* __restrict__ centers, const float* __restrict__ radii,
    const float* __restrict__ colors,  const float* __restrict__ lightP,
    const float* __restrict__ lightC,  float* __restrict__ out)
{
  __shared__ float  stage[STG_N];  // raw inputs, filled by async-to-LDS DMA
  __shared__ float4 sC[16];        // center.xyz, w = |c|^2 - r^2
  __shared__ float4 sCol[16];      // sphere color

  const int tid = threadIdx.x;

  // ---- Stage all inputs into LDS with CDNA5 async global->LDS loads ----
  if (tid < STG_N) {
    const float* src;
    if      (tid < STG_RAD) src = centers + tid;
    else if (tid < STG_COL) src = radii   + (tid - STG_RAD);
    else if (tid < STG_LP ) src = colors  + (tid - STG_COL);
    else if (tid < STG_LC ) src = lightP  + (tid - STG_LP);
    else                    src = lightC  + (tid - STG_LC);
    unsigned dst = (unsigned)(uintptr_t)&stage[tid];   // low 32 bits = LDS offset
    asm volatile("global_load_async_to_lds_b32 %0, %1, off"
                 :: "v"(dst), "v"(src) : "memory");
  }
  asm volatile("s_wait_asynccnt 0" ::: "memory");
  __syncthreads();

  // Derived per-sphere constants
  if (tid < 16) {
    float cx = stage[STG_CEN + 3*tid + 0];
    float cy = stage[STG_CEN + 3*tid + 1];
    float cz = stage[STG_CEN + 3*tid + 2];
    float r  = stage[STG_RAD + tid];
    sC[tid]   = make_float4(cx, cy, cz, cx*cx + cy*cy + cz*cz - r*r);
    sCol[tid] = make_float4(stage[STG_COL + 3*tid + 0],
                            stage[STG_COL + 3*tid + 1],
                            stage[STG_COL + 3*tid + 2], 0.0f);
  }
  __syncthreads();

  const int  lane  = tid & 31;
  const bool hi    = lane >= 16;        // upper half-wave
  const int  m     = lane & 15;
  const int  mbase = hi ? 8 : 0;

  // Loop-invariant A matrix: 16x4 sphere centers, K padded (cx,cy,cz,0).
  // Layout (32-bit A 16x4): lanes0-15 -> K=0,1 in V0,V1 ; lanes16-31 -> K=2,3.
  float4 cm = sC[m];
  v2f A;
  A.x = hi ? cm.z : cm.x;
  A.y = hi ? 0.0f : cm.y;

  // Per-lane sphere constants q = |c|^2 - r^2 for this lane's 8 D-rows
  float q[8];
  #pragma unroll
  for (int vv = 0; vv < 8; ++vv) q[vv] = sC[mbase + vv].w;

  const float lpx = stage[STG_LP+0], lpy = stage[STG_LP+1], lpz = stage[STG_LP+2];
  const float lcx = stage[STG_LC+0], lcy = stage[STG_LC+1], lcz = stage[STG_LC+2];
  const float INF = __builtin_inff();

  const int gid    = blockIdx.x * blockDim.x + threadIdx.x;
  const int stride = gridDim.x * blockDim.x;   // wave-uniform loop (NPIX divisible)

  for (int p = gid; p < NPIX; p += stride) {
    // Ray direction for THIS lane's pixel
    const int x = p & (WW - 1);
    const int y = p >> 10;
    const float u = -1.0f + 2.0f * (float)x * (1.0f/1023.0f);
    const float v =  1.0f - 2.0f * (float)y * (1.0f/1023.0f);
    const float inv = rsqrtf(u*u + 25.0f + v*v);
    const float dx = u*inv, dy = 5.0f*inv, dz = v*inv;

    // Other half-wave's direction (for building B of the other 16-ray group)
    const float ox = __shfl_xor(dx, 16, 32);
    const float oy = __shfl_xor(dy, 16, 32);
    const float oz = __shfl_xor(dz, 16, 32);

    // B matrices 4x16 (K x N): lanes0-15 hold K=0,1 ; lanes16-31 hold K=2,3.
    v2f B1, B2;
    B1.x = hi ? oz : dx;  B1.y = hi ? 0.0f : dy;
    B2.x = hi ? dz : ox;  B2.y = hi ? 0.0f : oy;

    v8f acc = {};
    // D[m][n] = c_m . d_n  (16 spheres x 16 rays)
    v8f D1 = __builtin_amdgcn_wmma_f32_16x16x4_f32(false, A, false, B1, (short)0, acc, false, false);
    v8f D2 = __builtin_amdgcn_wmma_f32_16x16x4_f32(false, A, false, B2, (short)0, acc, false, false);

    // Closest hit over this lane's 8 spheres, both ray groups — branchless:
    // raw v_sqrt_f32; sqrt(neg)=NaN is discarded by the disc>=0 select.
    float t1 = INF; int i1 = mbase;
    float t2 = INF; int i2 = mbase;
    #pragma unroll
    for (int vv = 0; vv < 8; ++vv) {
      float dc   = D1[vv];
      float disc = dc*dc - q[vv];
      float s    = __builtin_amdgcn_sqrtf(disc);
      float tc   = (disc >= 0.0f) ? (dc - s) : INF;
      if (tc < t1) { t1 = tc; i1 = mbase + vv; }
      dc   = D2[vv];
      disc = dc*dc - q[vv];
      s    = __builtin_amdgcn_sqrtf(disc);
      tc   = (disc >= 0.0f) ? (dc - s) : INF;
      if (tc < t2) { t2 = tc; i2 = mbase + vv; }
    }

    // Merge the two sphere halves (lane n <-> lane n+16), tie -> smaller index
    float t1o = __shfl_xor(t1, 16, 32); int i1o = __shfl_xor(i1, 16, 32);
    bool  take = (t1o < t1) || (t1o == t1 && i1o < i1);
    float T1 = take ? t1o : t1; int I1 = take ? i1o : i1;

    float t2o = __shfl_xor(t2, 16, 32); int i2o = __shfl_xor(i2, 16, 32);
    take = (t2o < t2) || (t2o == t2 && i2o < i2);
    float T2 = take ? t2o : t2; int I2 = take ? i2o : i2;

    const float T = hi ? T2 : T1;
    const int   I = hi ? I2 : I1;

    const bool hit = (T < INF);
    const float ts = hit ? T : 1.0f;

    // Phong shading (camera at origin -> view = -hitpoint)
    const float hx = ts*dx, hy = ts*dy, hz = ts*dz;
    const float4 sc = sC[I];
    float nx = hx - sc.x, ny = hy - sc.y, nz = hz - sc.z;
    const float ninv = rsqrtf(nx*nx + ny*ny + nz*nz);
    nx *= ninv; ny *= ninv; nz *= ninv;

    float lx = lpx - hx, ly = lpy - hy, lz = lpz - hz;
    const float linv = rsqrtf(lx*lx + ly*ly + lz*lz);
    lx *= linv; ly *= linv; lz *= linv;

    float vx = -hx, vy = -hy, vz = -hz;
    const float vinv = rsqrtf(vx*vx + vy*vy + vz*vz);
    vx *= vinv; vy *= vinv; vz *= vinv;

    const float ndl     = nx*lx + ny*ly + nz*lz;
    const float diffuse = 0.5f * fmaxf(ndl, 0.0f);
    const float rx = 2.0f*nx*ndl - lx;
    const float ry = 2.0f*ny*ndl - ly;
    const float rz = 2.0f*nz*ndl - lz;
    float rv = fmaxf(rx*vx + ry*vy + rz*vz, 0.0f);
    rv = rv*rv; rv = rv*rv; rv = rv*rv; rv = rv*rv;   // ^16
    const float k = 0.4f + diffuse + 0.8f*rv;

    const float4 scol = sCol[I];
    const float cr = fminf(fmaxf(k * scol.x * lcx, 0.0f), 1.0f);
    const float cg = fminf(fmaxf(k * scol.y * lcy, 0.0f), 1.0f);
    const float cb = fminf(fmaxf(k * scol.z * lcz, 0.0f), 1.0f);

    // Background gradient for row y
    const float vsb = -1.0f + 2.0f * (float)y * (1.0f/1023.0f);
    const float w1  = 1.0f - vsb;
    const float br  = fminf(fmaxf(w1*0.5f + vsb, 0.0f), 1.0f);
    const float bg  = fminf(fmaxf(w1*0.7f + vsb, 0.0f), 1.0f);
    const float bb  = 1.0f;

    float3 px;
    px.x = hit ? cr : br;
    px.y = hit ? cg : bg;
    px.z = hit ? cb : bb;
    ((float3*)out)[p] = px;   // single 12B (b96) store per pixel
  }
}

extern "C" void kernel_launch(void* const* d_in, const int* in_sizes, int n_in,
                              void* d_out, int out_size, void* d_ws, size_t ws_size,
                              hipStream_t stream) {
  (void)in_sizes; (void)n_in; (void)out_size; (void)d_ws; (void)ws_size;
  const float* centers = (const float*)d_in[0];
  const float* radii   = (const float*)d_in[1];
  const float* colors  = (const float*)d_in[2];
  const float* lightP  = (const float*)d_in[3];
  const float* lightC  = (const float*)d_in[4];
  float* out = (float*)d_out;

  dim3 block(256);
  dim3 grid(1024);   // 262144 threads -> 4 wave-uniform iterations each
  raytrace_wmma_kernel<<<grid, block, 0, stream>>>(centers, radii, colors, lightP, lightC, out);
}